// GINE_58171037057250
// MI455X (gfx1250) — compile-verified
//
#include <hip/hip_runtime.h>

#define DD 112    // hidden dim
#define GG 128    // num graphs
#define RPB 128   // GEMM rows per block
#define SWP (DD + 2)  // padded LDS row stride (even -> 8B-aligned v2f, breaks 2^k bank pattern)

typedef __attribute__((ext_vector_type(2))) float v2f;
typedef __attribute__((ext_vector_type(8))) float v8f;

// ---------------- utility kernels ----------------

__global__ void zero_f32(float* __restrict__ p, int n) {
  int g = blockIdx.x * blockDim.x + threadIdx.x;
  if (g < n) p[g] = 0.f;
}

__global__ void copy_f4(const float4* __restrict__ s, float4* __restrict__ d, int n4) {
  int g = blockIdx.x * blockDim.x + threadIdx.x;
  if (g < n4) d[g] = s[g];
}

// ---------------- encoders ----------------

// node_tables: (7, 258, 16) ; x: (N, 7) int32 ; h: (N, 112)
__global__ void node_enc(const int* __restrict__ x, const float* __restrict__ nt,
                         float* __restrict__ h, int n) {
  int g = blockIdx.x * blockDim.x + threadIdx.x;
  if (g >= n * DD) return;
  int node = g / DD;
  int d = g - node * DD;
  int i = d >> 4;                       // which of 7 tables (16 cols each)
  int xi = x[node * 7 + i];
  h[g] = nt[(i * 258 + xi) * 16 + (d & 15)];
}

// ---------------- edge message + scatter ----------------
// z must be pre-initialized to h (so z = h + sum relu(h[src]+e) after this).
// edge_tables: (2, 4, 56). e computed on the fly (tables are L1-resident).
// 28 threads per edge, each handles 4 contiguous floats (float4 gather of h[src]).
__global__ void edge_msg(const float* __restrict__ h, const int* __restrict__ ei,
                         const int* __restrict__ ea, const float* __restrict__ et,
                         float* __restrict__ z, int E) {
  int g = blockIdx.x * blockDim.x + threadIdx.x;
  int total = E * 28;
  if (g >= total) return;
  int e = g / 28;
  int part = g - e * 28;
  int d0 = part * 4;
  int src = ei[e];
  int dst = ei[E + e];
  int a0 = ea[e * 2 + 0];
  int a1 = ea[e * 2 + 1];
  float4 hv = *(const float4*)(h + (size_t)src * DD + d0);
  float ev[4];
#pragma unroll
  for (int j = 0; j < 4; ++j) {
    int d = d0 + j;
    ev[j] = (d < 56) ? et[a0 * 56 + d] : et[(4 + a1) * 56 + (d - 56)];
  }
  float m0 = fmaxf(hv.x + ev[0], 0.f);
  float m1 = fmaxf(hv.y + ev[1], 0.f);
  float m2 = fmaxf(hv.z + ev[2], 0.f);
  float m3 = fmaxf(hv.w + ev[3], 0.f);
  float* zp = z + (size_t)dst * DD + d0;
  atomicAdd(zp + 0, m0);
  atomicAdd(zp + 1, m1);
  atomicAdd(zp + 2, m2);
  atomicAdd(zp + 3, m3);
}

// ---------------- WMMA GEMM: Y[N,112] = A[N,112] @ W[112,112] + bias ----------------
// Block = 224 threads = 7 waves; block covers RPB=128 rows, wave w covers cols [16w,16w+16).
// W staged TRANSPOSED in LDS so each lane's B K-pair is one contiguous ds_load_b64.
// Per wave: 8 M-tiles x 28 V_WMMA_F32_16X16X4_F32.
__global__ __launch_bounds__(224) void gemm_wmma_f32(
    const float* __restrict__ A, const float* __restrict__ W,
    const float* __restrict__ bias, float* __restrict__ Y, int nrows) {
  __shared__ float sW[DD * SWP];    // sW[c*SWP + k] = W[k, c]   (51072 B)
  __shared__ float sA[RPB * SWP];   // sA[r*SWP + c] = A[row0+r, c] (58368 B)
  const int tid = threadIdx.x;
  // stage W transposed: coalesced global read, strided LDS write (once per block)
  for (int i = tid; i < DD * DD; i += 224) {
    int k = i / DD, c = i - k * DD;
    sW[c * SWP + k] = W[i];
  }
  const int row0 = blockIdx.x * RPB;
  for (int i = tid; i < RPB * DD; i += 224) {
    int r = i / DD, c = i - r * DD;
    int gr = row0 + r;
    sA[r * SWP + c] = (gr < nrows) ? A[(size_t)gr * DD + c] : 0.f;
  }
  __syncthreads();
  const int lane = tid & 31;
  const int col0 = (tid >> 5) * 16;     // wave -> column tile
  const int m = lane & 15;              // A row / B col within tile
  const int kq = (lane >> 4) << 1;      // lanes 0-15: K pair {0,1}; 16-31: {2,3}
  const int col = col0 + m;
  const float bb = bias[col];
  const float* bBase = &sW[col * SWP + kq];
  for (int mt = 0; mt < RPB; mt += 16) {
    const float* aBase = &sA[(mt + m) * SWP + kq];
    v8f acc = {};
#pragma unroll
    for (int k = 0; k < DD; k += 4) {
      v2f a = *(const v2f*)(aBase + k);
      v2f b = *(const v2f*)(bBase + k);
      acc = __builtin_amdgcn_wmma_f32_16x16x4_f32(false, a, false, b,
                                                  (short)0, acc, false, false);
    }
    // C/D layout: VGPR r -> row r (lanes 0-15) / row 8+r (lanes 16-31), col = lane&15
    const int rbase = mt + ((lane >> 4) * 8);
#pragma unroll
    for (int r = 0; r < 8; ++r) {
      int gr = row0 + rbase + r;
      if (gr < nrows) Y[(size_t)gr * DD + col] = acc[r] + bb;
    }
  }
}

// ---------------- column statistics (sum, sumsq) ----------------
// blockDim must be a multiple of cols (224 works for 112 / 56 / 28).
__global__ void col_stats(const float* __restrict__ y, float* __restrict__ stats,
                          int n, int cols) {
  int tid = threadIdx.x;
  int col = tid % cols;
  int sub = tid / cols;
  int rpb = blockDim.x / cols;
  float s = 0.f, ss = 0.f;
  for (int r = blockIdx.x * rpb + sub; r < n; r += gridDim.x * rpb) {
    float v = y[(size_t)r * cols + col];
    s += v;
    ss += v * v;
  }
  atomicAdd(&stats[col], s);
  atomicAdd(&stats[cols + col], ss);
}

// out = relu(g*(y-mu)*rsqrt(var+eps)+be) ; mu/var from stats (biased, training mode)
__global__ void bn_relu(const float* __restrict__ y, const float* __restrict__ stats,
                        const float* __restrict__ gmm, const float* __restrict__ beta,
                        float* __restrict__ out, int n, int cols) {
  int g = blockIdx.x * blockDim.x + threadIdx.x;
  if (g >= n * cols) return;
  int col = g % cols;
  float inv_n = 1.f / (float)n;
  float mu = stats[col] * inv_n;
  float var = stats[cols + col] * inv_n - mu * mu;
  float r = rsqrtf(var + 1e-5f);
  float v = gmm[col] * (y[g] - mu) * r + beta[col];
  out[g] = fmaxf(v, 0.f);
}

// ---------------- pooling + MLP head ----------------

__global__ void pool_sum(const float* __restrict__ h, const int* __restrict__ batch,
                         float* __restrict__ p, int n) {
  int g = blockIdx.x * blockDim.x + threadIdx.x;
  if (g >= n * DD) return;
  int node = g / DD;
  int d = g - node * DD;
  atomicAdd(&p[batch[node] * DD + d], h[g]);
}

__global__ void small_gemm(const float* __restrict__ A, const float* __restrict__ W,
                           const float* __restrict__ b, float* __restrict__ Y,
                           int mrows, int k, int ncols) {
  int g = blockIdx.x * blockDim.x + threadIdx.x;
  if (g >= mrows * ncols) return;
  int row = g / ncols, col = g - row * ncols;
  float acc = b[col];
  for (int i = 0; i < k; ++i) acc += A[row * k + i] * W[i * ncols + col];
  Y[g] = acc;
}

// ---------------- host orchestration ----------------

static inline int cdiv(long long a, long long b) { return (int)((a + b - 1) / b); }

extern "C" void kernel_launch(void* const* d_in, const int* in_sizes, int n_in,
                              void* d_out, int out_size, void* d_ws, size_t ws_size,
                              hipStream_t stream) {
  const int*   x           = (const int*)d_in[0];
  const int*   edge_index  = (const int*)d_in[1];
  const int*   edge_attr   = (const int*)d_in[2];
  const int*   batch       = (const int*)d_in[3];
  const float* node_tables = (const float*)d_in[4];
  const float* edge_tables = (const float*)d_in[5];
  const float* conv_W1     = (const float*)d_in[6];
  const float* conv_b1     = (const float*)d_in[7];
  const float* conv_g1     = (const float*)d_in[8];
  const float* conv_be1    = (const float*)d_in[9];
  const float* conv_W2     = (const float*)d_in[10];
  const float* conv_b2     = (const float*)d_in[11];
  const float* conv_g2     = (const float*)d_in[12];
  const float* conv_be2    = (const float*)d_in[13];
  const float* mlp_W1      = (const float*)d_in[14];
  const float* mlp_b1      = (const float*)d_in[15];
  const float* mlp_g1      = (const float*)d_in[16];
  const float* mlp_be1     = (const float*)d_in[17];
  const float* mlp_W2      = (const float*)d_in[18];
  const float* mlp_b2      = (const float*)d_in[19];
  const float* mlp_g2      = (const float*)d_in[20];
  const float* mlp_be2     = (const float*)d_in[21];
  const float* mlp_W3      = (const float*)d_in[22];
  const float* mlp_b3      = (const float*)d_in[23];
  float* out = (float*)d_out;

  const int N = in_sizes[3];          // nodes
  const int E = in_sizes[1] / 2;      // edges

  float* ws    = (float*)d_ws;
  float* h     = ws;                          // N*DD
  float* z     = h + (size_t)N * DD;          // N*DD
  float* y     = z + (size_t)N * DD;          // N*DD
  float* stats = y + (size_t)N * DD;          // 256
  float* p     = stats + 256;                 // GG*DD
  float* q     = p + GG * DD;                 // GG*56
  float* rbuf  = q + GG * 56;                 // GG*28

  // encoders
  node_enc<<<cdiv((long long)N * DD, 256), 256, 0, stream>>>(x, node_tables, h, N);

  for (int l = 0; l < 3; ++l) {
    const float* W1  = conv_W1  + (size_t)l * DD * DD;
    const float* b1  = conv_b1  + (size_t)l * DD;
    const float* g1  = conv_g1  + (size_t)l * DD;
    const float* be1 = conv_be1 + (size_t)l * DD;
    const float* W2  = conv_W2  + (size_t)l * DD * DD;
    const float* b2  = conv_b2  + (size_t)l * DD;
    const float* g2  = conv_g2  + (size_t)l * DD;
    const float* be2 = conv_be2 + (size_t)l * DD;

    // z = h, then z += segment_sum(relu(h[src]+e), dst)
    copy_f4<<<cdiv((long long)N * DD / 4, 256), 256, 0, stream>>>(
        (const float4*)h, (float4*)z, N * DD / 4);
    edge_msg<<<cdiv((long long)E * 28, 256), 256, 0, stream>>>(
        h, edge_index, edge_attr, edge_tables, z, E);

    // y = z @ W1 + b1 ; h = relu(bn(y))
    gemm_wmma_f32<<<cdiv(N, RPB), 224, 0, stream>>>(z, W1, b1, y, N);
    zero_f32<<<1, 256, 0, stream>>>(stats, 2 * DD);
    col_stats<<<256, 224, 0, stream>>>(y, stats, N, DD);
    bn_relu<<<cdiv((long long)N * DD, 256), 256, 0, stream>>>(y, stats, g1, be1, h, N, DD);

    // y = h @ W2 + b2 ; h = relu(bn(y))
    gemm_wmma_f32<<<cdiv(N, RPB), 224, 0, stream>>>(h, W2, b2, y, N);
    zero_f32<<<1, 256, 0, stream>>>(stats, 2 * DD);
    col_stats<<<256, 224, 0, stream>>>(y, stats, N, DD);
    bn_relu<<<cdiv((long long)N * DD, 256), 256, 0, stream>>>(y, stats, g2, be2, h, N, DD);
  }

  // global add pool -> p[G,112]
  zero_f32<<<cdiv(GG * DD, 256), 256, 0, stream>>>(p, GG * DD);
  pool_sum<<<cdiv((long long)N * DD, 256), 256, 0, stream>>>(h, batch, p, N);

  // MLP head: 112 -> 56 -> 28 -> 1 (BN over G=128 rows)
  small_gemm<<<cdiv(GG * 56, 256), 256, 0, stream>>>(p, mlp_W1, mlp_b1, q, GG, DD, 56);
  zero_f32<<<1, 256, 0, stream>>>(stats, 2 * 56);
  col_stats<<<8, 224, 0, stream>>>(q, stats, GG, 56);
  bn_relu<<<cdiv(GG * 56, 256), 256, 0, stream>>>(q, stats, mlp_g1, mlp_be1, q, GG, 56);

  small_gemm<<<cdiv(GG * 28, 256), 256, 0, stream>>>(q, mlp_W2, mlp_b2, rbuf, GG, 56, 28);
  zero_f32<<<1, 256, 0, stream>>>(stats, 2 * 28);
  col_stats<<<4, 224, 0, stream>>>(rbuf, stats, GG, 28);
  bn_relu<<<cdiv(GG * 28, 256), 256, 0, stream>>>(rbuf, stats, mlp_g2, mlp_be2, rbuf, GG, 28);

  small_gemm<<<cdiv(GG, 256), 256, 0, stream>>>(rbuf, mlp_W3, mlp_b3, out, GG, 28, 1);

  (void)n_in; (void)out_size; (void)ws_size;
}